// GATrAutoRegressorLoss_5858335392440
// MI455X (gfx1250) — compile-verified
//
#include <hip/hip_runtime.h>
#include <math.h>

#define TT   32
#define BB   256
#define NN   500000
#define NPF  4096

#define L_DIR 1.0f
#define L_MAG 1.0f
#define L_PID 1.0f
#define L_CHG 0.5f
#define L_ASN 1.0f
#define L_STP 0.5f

typedef __attribute__((ext_vector_type(2))) float v2f;
typedef __attribute__((ext_vector_type(8))) float v8f;

struct Acc {
    double dir_sum, mag_sum, pid_sum, chg_sum, vcnt;
    double asn_sum, acnt, stop_sum;
    int    ppe[BB];
};

// ---------------------------------------------------------------------------
// Wave32 full reduction via V_WMMA_F32_16X16X4_F32.
// A: 16x4 f32 (2 VGPRs). VGPR0: lanes0-15 -> A[m][0], lanes16-31 -> A[m][2].
// With a.x = val, a.y = 0, B = ones(4x16):  D[m][n] = val[m] + val[m+16].
// C/D layout: VGPR r: lanes0-15 = D[r][lane], lanes16-31 = D[8+r][lane-16].
// Sum of the 8 D VGPRs per lane gives half-sums; xor-16 shuffle completes it.
// EXEC must be all ones at call sites (callers guarantee reconvergence).
// ---------------------------------------------------------------------------
__device__ __forceinline__ float wave_sum32(float val) {
    v2f a; a.x = val;  a.y = 0.0f;
    v2f b; b.x = 1.0f; b.y = 1.0f;
    v8f c = {};
    c = __builtin_amdgcn_wmma_f32_16x16x4_f32(false, a, false, b,
                                              (short)0, c, false, false);
    float p = c[0] + c[1] + c[2] + c[3] + c[4] + c[5] + c[6] + c[7];
    p += __shfl_xor(p, 16, 32);
    return p;  // total in every lane
}

__device__ __forceinline__ float softplusf(float x) {
    // logaddexp(0, x) = max(x,0) + log1p(exp(-|x|)), numerically stable
    return fmaxf(x, 0.0f) + log1pf(__expf(-fabsf(x)));
}

// ---------------------------------------------------------------------------
__global__ void k_zero(Acc* __restrict__ acc) {
    int i = threadIdx.x;
    if (i == 0) {
        acc->dir_sum = 0.0; acc->mag_sum = 0.0; acc->pid_sum = 0.0;
        acc->chg_sum = 0.0; acc->vcnt = 0.0;
        acc->asn_sum = 0.0; acc->acnt = 0.0; acc->stop_sum = 0.0;
    }
    if (i < BB) acc->ppe[i] = 0;
}

__global__ void k_ppe(const int* __restrict__ gt_batch, Acc* __restrict__ acc) {
    int i = blockIdx.x * blockDim.x + threadIdx.x;
    if (i < NPF) atomicAdd(&acc->ppe[gt_batch[i]], 1);
}

// ---------------------------------------------------------------------------
// Per-PFO losses (dir/mag/pid/charge). gt_batch is sorted, so the step index
// within an event is i - lower_bound(gt_batch, gt_batch[i]). PFOs with
// step < T are exactly the valid (t,b) grid positions of the reference.
// ---------------------------------------------------------------------------
__global__ void k_pfo(const float* __restrict__ pm,   const float* __restrict__ ppd,
                      const float* __restrict__ ppid, const float* __restrict__ pchg,
                      const float* __restrict__ gm,   const float* __restrict__ gpd,
                      const float* __restrict__ gpid, const float* __restrict__ gchg,
                      const int*   __restrict__ gt_batch, Acc* __restrict__ acc) {
    int i = blockIdx.x * blockDim.x + threadIdx.x;   // 0..NPF-1 exactly
    float ldir = 0.f, lmag = 0.f, lpid = 0.f, lchg = 0.f, lcnt = 0.f;

    int bb = gt_batch[i];
    int lo = 0, hi = NPF;
    while (lo < hi) { int mid = (lo + hi) >> 1; if (gt_batch[mid] < bb) lo = mid + 1; else hi = mid; }
    int s = i - lo;   // within-event step index

    if (s < TT) {
        int pb = s * BB + bb;
        // direction: 1 - cosine similarity with eps-clamped norms
        float ax = pm[pb*3+0], ay = pm[pb*3+1], az = pm[pb*3+2];
        float gx = gm[i*3+0],  gy = gm[i*3+1],  gz = gm[i*3+2];
        float na = fmaxf(sqrtf(ax*ax + ay*ay + az*az), 1e-8f);
        float ng = fmaxf(sqrtf(gx*gx + gy*gy + gz*gz), 1e-8f);
        ldir = 1.0f - (ax*gx + ay*gy + az*gz) / (na * ng);
        // magnitude MSE
        float dm = ppd[pb] - gpd[i];
        lmag = dm * dm;
        // PID: NLL at argmax(gt_pid) (first max, matching jnp.argmax)
        int cls = 0; float bv = gpid[i*5+0];
        #pragma unroll
        for (int c = 1; c < 5; ++c) { float v = gpid[i*5+c]; if (v > bv) { bv = v; cls = c; } }
        float xv[5]; float mx = -3.4e38f;
        #pragma unroll
        for (int c = 0; c < 5; ++c) { xv[c] = ppid[pb*5+c]; mx = fmaxf(mx, xv[c]); }
        float se = 0.f;
        #pragma unroll
        for (int c = 0; c < 5; ++c) se += __expf(xv[c] - mx);
        lpid = (mx + __logf(se)) - xv[cls];
        // charge MSE
        float dc = pchg[pb] - gchg[i];
        lchg = dc * dc;
        lcnt = 1.0f;
    }

    float sdir = wave_sum32(ldir);
    float smag = wave_sum32(lmag);
    float spid = wave_sum32(lpid);
    float schg = wave_sum32(lchg);
    float scnt = wave_sum32(lcnt);
    if ((threadIdx.x & 31) == 0) {
        atomicAdd(&acc->dir_sum, (double)sdir);
        atomicAdd(&acc->mag_sum, (double)smag);
        atomicAdd(&acc->pid_sum, (double)spid);
        atomicAdd(&acc->chg_sum, (double)schg);
        atomicAdd(&acc->vcnt,    (double)scnt);
    }
}

// ---------------------------------------------------------------------------
__global__ void k_stop(const float* __restrict__ stop_logits, Acc* __restrict__ acc) {
    int i = blockIdx.x * blockDim.x + threadIdx.x;   // 0..T*B-1 exactly
    int t = i >> 8, b = i & (BB - 1);
    float x = stop_logits[i];
    float z = (t >= acc->ppe[b]) ? 1.0f : 0.0f;
    float l = softplusf(x) - x * z;
    float s = wave_sum32(l);
    if ((threadIdx.x & 31) == 0) atomicAdd(&acc->stop_sum, (double)s);
}

// ---------------------------------------------------------------------------
// Assignment BCE: the heavy kernel. Thread n walks column n of the (T,N)
// logits with stride N; only t < min(ppe[hit_batch[n]],T) contribute (the
// rest are masked out in the reference, so we skip the loads entirely).
// hit_batch is sorted -> neighboring lanes share v -> coalesced, uniform loops.
// ---------------------------------------------------------------------------
__global__ void k_assign(const float* __restrict__ alog,
                         const int*   __restrict__ hit_to_pfo,
                         const int*   __restrict__ hit_batch,
                         Acc* __restrict__ acc) {
    int n = blockIdx.x * blockDim.x + threadIdx.x;
    float lsum = 0.f, lcnt = 0.f;
    if (n < NN) {
        int v = acc->ppe[hit_batch[n]];
        v = v < TT ? v : TT;
        int h = hit_to_pfo[n];           // in [0, T)
        lcnt = (float)v;
        const float* col = alog + n;
        __builtin_prefetch(col, 0, 0);   // global_prefetch_b8
        for (int t = 0; t < v; ++t) {
            if (t + 4 < v) __builtin_prefetch(col + (size_t)(t + 4) * NN, 0, 0);
            float x = col[(size_t)t * NN];
            lsum += softplusf(x);        // bce(x,0)
            if (t == h) lsum -= x;       // bce(x,1) = softplus(x) - x
        }
    }
    float s = wave_sum32(lsum);
    float c = wave_sum32(lcnt);
    if ((threadIdx.x & 31) == 0) {
        atomicAdd(&acc->asn_sum, (double)s);
        atomicAdd(&acc->acnt,    (double)c);
    }
}

// ---------------------------------------------------------------------------
__global__ void k_final(const Acc* __restrict__ acc, float* __restrict__ out) {
    if (blockIdx.x == 0 && threadIdx.x == 0) {
        double vc = acc->vcnt > 1.0 ? acc->vcnt : 1.0;
        double ac = acc->acnt > 1.0 ? acc->acnt : 1.0;
        float dir = (float)(acc->dir_sum / vc);
        float mag = (float)(acc->mag_sum / vc);
        float pid = (float)(acc->pid_sum / vc);
        float chg = (float)(acc->chg_sum / vc);
        float asn = (float)(acc->asn_sum / ac);
        float stp = (float)(acc->stop_sum / (double)(TT * BB));
        float total = L_DIR*dir + L_MAG*mag + L_PID*pid + L_CHG*chg + L_ASN*asn + L_STP*stp;
        out[0] = total; out[1] = dir; out[2] = mag; out[3] = pid;
        out[4] = chg;   out[5] = asn; out[6] = stp;
    }
}

// ---------------------------------------------------------------------------
extern "C" void kernel_launch(void* const* d_in, const int* in_sizes, int n_in,
                              void* d_out, int out_size, void* d_ws, size_t ws_size,
                              hipStream_t stream) {
    const float* pfo_momentum = (const float*)d_in[0];   // (T,B,3)
    const float* pfo_p_mod    = (const float*)d_in[1];   // (T,B,1)
    const float* pfo_pid      = (const float*)d_in[2];   // (T,B,5)
    const float* pfo_charge   = (const float*)d_in[3];   // (T,B,1)
    const float* asn_logits   = (const float*)d_in[4];   // (T,N,1)
    const float* stop_logits  = (const float*)d_in[5];   // (T,B,1)
    const float* gt_momentum  = (const float*)d_in[6];   // (NPFO,3)
    const float* gt_p_mod     = (const float*)d_in[7];   // (NPFO,1)
    const float* gt_pid       = (const float*)d_in[8];   // (NPFO,5)
    const float* gt_charge    = (const float*)d_in[9];   // (NPFO,1)
    const int*   gt_batch     = (const int*)d_in[10];    // (NPFO,) sorted
    const int*   hit_to_pfo   = (const int*)d_in[11];    // (N,)
    const int*   hit_batch    = (const int*)d_in[12];    // (N,) sorted

    Acc*   acc = (Acc*)d_ws;
    float* out = (float*)d_out;

    k_zero<<<1, 256, 0, stream>>>(acc);
    k_ppe<<<NPF / 256, 256, 0, stream>>>(gt_batch, acc);
    k_pfo<<<NPF / 256, 256, 0, stream>>>(pfo_momentum, pfo_p_mod, pfo_pid, pfo_charge,
                                         gt_momentum, gt_p_mod, gt_pid, gt_charge,
                                         gt_batch, acc);
    k_stop<<<(TT * BB) / 256, 256, 0, stream>>>(stop_logits, acc);
    k_assign<<<(NN + 255) / 256, 256, 0, stream>>>(asn_logits, hit_to_pfo, hit_batch, acc);
    k_final<<<1, 64, 0, stream>>>(acc, out);
}